// OneDtoTwoDLayer_51092930953908
// MI455X (gfx1250) — compile-verified
//
#include <hip/hip_runtime.h>
#include <stdint.h>

// out[b,i,d] = relu(x[b,i] * W[i,d] + bias[i,d]),  B=256, S=512, D=512, all f32.
//
// Roofline: 0.13 GFLOP vs 268 MB moved (256 MB stores) => ~0.5 FLOP/byte.
// Pure HBM-store-bound; floor ~= 268 MB / 23.3 TB/s ~= 11.5 us on MI455X.
// Design: async-stage W row / bias row / x column into LDS (ASYNCcnt path),
// compute each 16x16 (b,d) tile with one V_WMMA_F32_16X16X4_F32 rank-1 update
// (only K=0 populated), ReLU via v_max, stores follow the WMMA C/D layout
// (VGPR j -> rows j/j+8, lanes -> N) giving 64B-contiguous segments.
// v2: all LDS reads + lane selects hoisted out of the tile loop; 16 tiles per
// wave fully unrolled; stores use immediate offsets (j*1MB + q*64B < 8MB).

namespace {
constexpr int kB       = 256;
constexpr int kS       = 512;
constexpr int kD       = 512;
constexpr int kBChunk  = 64;    // batch rows per block
constexpr int kThreads = 256;   // 8 wave32
}

typedef __attribute__((ext_vector_type(2))) float v2f;
typedef __attribute__((ext_vector_type(8))) float v8f;

__global__ __launch_bounds__(kThreads)
void onedtotwod_wmma_kernel(const float* __restrict__ X,     // [B,S]
                            const float* __restrict__ W,     // [S,D]
                            const float* __restrict__ Bias,  // [S,D]
                            float* __restrict__ Out)         // [B,S,D]
{
    __shared__ __align__(16) float sW[kD];       // W[i, :]
    __shared__ __align__(16) float sB[kD];       // bias[i, :]
    __shared__ __align__(16) float sX[kBChunk];  // x[b0:b0+64, i]

    const int i  = blockIdx.x >> 2;          // sequence position 0..511
    const int b0 = (blockIdx.x & 3) * kBChunk;
    const int t  = threadIdx.x;

    // ---- Async stage parameters into LDS (GLOBAL_LOAD_ASYNC_TO_LDS) ----
    {
        // 256 threads x 16B = 4KB: t<128 -> W row, t>=128 -> bias row.
        const float* g16 = (t < 128) ? (W + i * kD + t * 4)
                                     : (Bias + i * kD + (t - 128) * 4);
        unsigned l16 = (t < 128) ? (unsigned)(uintptr_t)&sW[t * 4]
                                 : (unsigned)(uintptr_t)&sB[(t - 128) * 4];
        unsigned long long ga = (unsigned long long)(uintptr_t)g16;
        asm volatile("global_load_async_to_lds_b128 %0, %1, off"
                     :: "v"(l16), "v"(ga) : "memory");

        if (t < kBChunk) {  // 64 strided scalars of x (stride S floats)
            const float* gx = X + (b0 + t) * kS + i;
            unsigned lx = (unsigned)(uintptr_t)&sX[t];
            unsigned long long gxa = (unsigned long long)(uintptr_t)gx;
            asm volatile("global_load_async_to_lds_b32 %0, %1, off"
                         :: "v"(lx), "v"(gxa) : "memory");
        }
    }
#if __has_builtin(__builtin_amdgcn_s_wait_asynccnt)
    __builtin_amdgcn_s_wait_asynccnt(0);
#else
    asm volatile("s_wait_asynccnt 0" ::: "memory");
#endif
    __syncthreads();

    const int lane = t & 31;
    const int wv   = t >> 5;      // wave id 0..7
    const int n    = lane & 15;   // N coordinate within a 16x16 tile
    const int half = lane >> 4;   // 0: lanes 0-15, 1: lanes 16-31
    const int d0w  = wv * 64;     // this wave's fixed 64-wide d slice

    // ---- Hoisted LDS reads (unconditional -> plain ds_load, no branches) ----
    float wq[4], bq[4], xa[4];
    #pragma unroll
    for (int q = 0; q < 4; ++q) {
        wq[q] = sW[d0w + q * 16 + n];
        bq[q] = sB[d0w + q * 16 + n];
    }
    #pragma unroll
    for (int bt = 0; bt < 4; ++bt) xa[bt] = sX[bt * 16 + n];

    // K=0-only fragments: A vgpr0 lanes 0-15 = (M=n, K=0), lanes 16-31 = K=2
    // slot (zero). One v_cndmask each, hoisted.
    float bmx[4], axv[4];
    #pragma unroll
    for (int q = 0; q < 4; ++q)  bmx[q] = (half == 0) ? wq[q] : 0.0f;
    #pragma unroll
    for (int bt = 0; bt < 4; ++bt) axv[bt] = (half == 0) ? xa[bt] : 0.0f;

    // ---- 16 tiles per wave: 4 b-tiles x 4 d-tiles, fully unrolled ----
    #pragma unroll
    for (int bt = 0; bt < 4; ++bt) {
        // One base pointer per b-tile; everything else is immediate offsets:
        //   q*16 floats = 64B,  j * S*D floats = j*1MB  (max 7.34MB < 8MB imm)
        float* base = Out + ((b0 + bt * 16 + half * 8) * kS + i) * kD + d0w + n;

        v2f a;
        a.x = axv[bt];
        a.y = 0.0f;

        #pragma unroll
        for (int q = 0; q < 4; ++q) {
            v2f bm;
            bm.x = bmx[q];
            bm.y = 0.0f;

            v8f c;  // C: bias broadcast along M
            #pragma unroll
            for (int j = 0; j < 8; ++j) c[j] = bq[q];

#if __has_builtin(__builtin_amdgcn_wmma_f32_16x16x4_f32)
            c = __builtin_amdgcn_wmma_f32_16x16x4_f32(
                    /*neg_a=*/false, a, /*neg_b=*/false, bm,
                    /*c_mod=*/(short)0, c, /*reuse_a=*/false, /*reuse_b=*/false);
#else
            #pragma unroll
            for (int j = 0; j < 8; ++j)
                c[j] = fmaf(xa[bt], wq[q], bq[q]);  // fallback (compile-only)
#endif

            #pragma unroll
            for (int j = 0; j < 8; ++j) {
                float v = c[j];
                base[q * 16 + j * kS * kD] = (v > 0.0f) ? v : 0.0f;  // ReLU
            }
        }
    }
}

extern "C" void kernel_launch(void* const* d_in, const int* in_sizes, int n_in,
                              void* d_out, int out_size, void* d_ws, size_t ws_size,
                              hipStream_t stream) {
    const float* X    = (const float*)d_in[0];
    const float* W    = (const float*)d_in[1];
    const float* Bias = (const float*)d_in[2];
    float* Out        = (float*)d_out;

    dim3 grid(kS * (kB / kBChunk));  // 512 * 4 = 2048 blocks
    dim3 block(kThreads);            // 256 threads = 8 wave32
    hipLaunchKernelGGL(onedtotwod_wmma_kernel, grid, block, 0, stream,
                       X, W, Bias, Out);
}